// S4_48627619725767
// MI455X (gfx1250) — compile-verified
//
#include <hip/hip_runtime.h>

// ---------------------------------------------------------------------------
// Mamba selective-scan forward for MI455X (gfx1250, wave32, WMMA).
//   B=2, L=2048, D_IN=2048, N=16, DT_RANK=128
// Pipeline:
//   1) gemm_xproj : bcd[m,0:160] = x[m,:] . W_xproj^T        (v_wmma_f32_16x16x4_f32)
//   2) gemm_dt    : delta[m,d]   = softplus(dt . W_dt^T + b) (v_wmma_f32_16x16x4_f32)
//   3) scan_phase1: per-chunk local scan -> P (prod of a), H (local end state)
//                   B-tile staged via global_load_async_to_lds_b128 (ASYNCcnt)
//   4) scan_phase2: inter-chunk recurrence -> Hinit (chunk initial states)
//   5) scan_phase3: re-scan with Hinit, y = sum_n h*C + x*D (B+C tiles async->LDS)
// ---------------------------------------------------------------------------

typedef float v2f __attribute__((ext_vector_type(2)));
typedef float v8f __attribute__((ext_vector_type(8)));

#define B_SZ     2
#define L_SZ     2048
#define D_SZ     2048
#define N_ST     16
#define R_SZ     128
#define E_SZ     160            // dt_rank + 2*N
#define CHUNK    64
#define NCHUNK   (L_SZ / CHUNK) // 32
#define M_SZ     (B_SZ * L_SZ)  // 4096

static __device__ __forceinline__ v8f wmma_k4(v2f a, v2f b, v8f c) {
  // D = A(16x4,f32) * B(4x16,f32) + C(16x16,f32)
  return __builtin_amdgcn_wmma_f32_16x16x4_f32(
      /*neg_a=*/false, a, /*neg_b=*/false, b,
      /*c_mod=*/(short)0, c, /*reuse_a=*/false, /*reuse_b=*/false);
}

// Async copy of 16 bytes global -> LDS, tracked by ASYNCcnt.
// lds_addr is the wave-relative LDS byte address (low 32 bits of the generic
// pointer to a __shared__ object, per the flat-address LDS truncation rule).
static __device__ __forceinline__ void async_g2lds_b128(const float* gsrc,
                                                        unsigned lds_addr) {
  asm volatile("global_load_async_to_lds_b128 %0, %1, off"
               :: "v"(lds_addr), "v"(gsrc)
               : "memory");
}
static __device__ __forceinline__ void wait_async0() {
  asm volatile("s_wait_asynccnt 0x0" ::: "memory");
}

// ---------------------------------------------------------------------------
// GEMM-1: bcd[m,e] = sum_k x[m,k] * W_xproj[e,k]   (M=4096, N=160, K=2048)
// One wave per 16x16 output tile. 2560 tiles, 8 waves/block -> 320 blocks.
// ---------------------------------------------------------------------------
__global__ void S4_gemm_xproj(const float* __restrict__ x,
                              const float* __restrict__ W,
                              float* __restrict__ bcd) {
  const int lane = threadIdx.x & 31;
  const int wave = threadIdx.x >> 5;
  const int tile = blockIdx.x * 8 + wave;
  if (tile >= (M_SZ / 16) * (E_SZ / 16)) return;
  const int mTile = tile / (E_SZ / 16);
  const int nTile = tile % (E_SZ / 16);
  const int mBase = mTile * 16, nBase = nTile * 16;

  const int lm = lane & 15;          // M (A-frag) / N (B-frag) index
  const int kh = (lane >> 4) * 2;    // K-pair selector

  const float* __restrict__ aRow = x + (size_t)(mBase + lm) * D_SZ + kh;
  const float* __restrict__ bRow = W + (size_t)(nBase + lm) * D_SZ + kh;

  v8f acc = {};
#pragma unroll 8
  for (int k = 0; k < D_SZ; k += 4) {
    v2f av = *(const v2f*)(aRow + k);
    v2f bv = *(const v2f*)(bRow + k);
    acc = wmma_k4(av, bv, acc);
  }
  // D layout: VGPR i -> M = mBase + i + (lane>>4)*8 ; N = nBase + (lane&15)
#pragma unroll
  for (int i = 0; i < 8; ++i) {
    int m = mBase + i + (lane >> 4) * 8;
    bcd[(size_t)m * E_SZ + nBase + lm] = acc[i];
  }
}

// ---------------------------------------------------------------------------
// GEMM-2: delta[m,d] = softplus( sum_r bcd[m,r] * W_dt[d,r] + b_dt[d] )
// M=4096, N=2048, K=128. 32768 tiles, 8 waves/block -> 4096 blocks.
// ---------------------------------------------------------------------------
__global__ void S4_gemm_dt(const float* __restrict__ bcd,
                           const float* __restrict__ Wdt,
                           const float* __restrict__ bdt,
                           float* __restrict__ delta) {
  const int lane = threadIdx.x & 31;
  const int wave = threadIdx.x >> 5;
  const int tile = blockIdx.x * 8 + wave;
  const int mTile = tile / (D_SZ / 16);
  const int nTile = tile % (D_SZ / 16);
  const int mBase = mTile * 16, nBase = nTile * 16;

  const int lm = lane & 15;
  const int kh = (lane >> 4) * 2;

  const float* __restrict__ aRow = bcd + (size_t)(mBase + lm) * E_SZ + kh;
  const float* __restrict__ bRow = Wdt + (size_t)(nBase + lm) * R_SZ + kh;

  v8f acc = {};
#pragma unroll 8
  for (int k = 0; k < R_SZ; k += 4) {
    v2f av = *(const v2f*)(aRow + k);
    v2f bv = *(const v2f*)(bRow + k);
    acc = wmma_k4(av, bv, acc);
  }
#pragma unroll
  for (int i = 0; i < 8; ++i) {
    int m = mBase + i + (lane >> 4) * 8;
    int d = nBase + lm;
    float z = acc[i] + bdt[d];
    float sp = (z > 20.0f) ? z : __logf(1.0f + __expf(z));  // softplus
    delta[(size_t)m * D_SZ + d] = sp;
  }
}

// ---------------------------------------------------------------------------
// Scan phase 1: per-(b,d,chunk) local scan with h0=0.
//   a_t = exp(delta*A[n]);  h = a*h + delta*B_t[n]*x
//   P[b,d,c,n] = prod a_t   ;  H[b,d,c,n] = local end state
// One thread per (b,d,chunk); block = 256 consecutive d of one (b,c).
// B tile (CHUNK x 16 floats, rows 16B-aligned) staged via async b128 -> LDS.
// ---------------------------------------------------------------------------
__global__ void S4_scan_phase1(const float* __restrict__ x,
                               const float* __restrict__ A,
                               const float* __restrict__ bcd,
                               const float* __restrict__ delta,
                               float* __restrict__ P,
                               float* __restrict__ H) {
  __shared__ float sB[CHUNK * N_ST];       // 4 KB: B values, shared over d
  const int idx = blockIdx.x * blockDim.x + threadIdx.x;
  const int d = idx & (D_SZ - 1);
  const int t = idx >> 11;
  const int b = t & (B_SZ - 1);
  const int c = t >> 1;

  const size_t rowBase = (size_t)b * L_SZ + (size_t)c * CHUNK;  // m index base
  {
    // 256 threads x 16B = 4KB: thread j stages quarter q of row l.
    const int l = threadIdx.x >> 2, q = threadIdx.x & 3;
    const float* gsrc = bcd + (rowBase + l) * E_SZ + R_SZ + q * 4;
    const unsigned ldst = (unsigned)(uintptr_t)(sB + l * N_ST + q * 4);
    async_g2lds_b128(gsrc, ldst);
    wait_async0();
  }
  __syncthreads();

  float Ar[N_ST];
#pragma unroll
  for (int n = 0; n < N_ST; ++n) Ar[n] = A[d * N_ST + n];

  float h[N_ST], p[N_ST];
#pragma unroll
  for (int n = 0; n < N_ST; ++n) { h[n] = 0.0f; p[n] = 1.0f; }

  for (int li = 0; li < CHUNK; ++li) {
    const size_t off = (rowBase + li) * D_SZ + d;
    float dlt = delta[off];
    float xv  = x[off];
    __builtin_prefetch(delta + off + D_SZ, 0, 0);   // global_prefetch_b8
    float dx = dlt * xv;
#pragma unroll
    for (int n = 0; n < N_ST; ++n) {
      float a = __expf(dlt * Ar[n]);
      h[n] = __builtin_fmaf(a, h[n], dx * sB[(li << 4) + n]);
      p[n] *= a;
    }
  }
  const size_t pidx = (((size_t)b * D_SZ + d) * NCHUNK + c) * N_ST;
#pragma unroll
  for (int n = 0; n < N_ST; ++n) { P[pidx + n] = p[n]; H[pidx + n] = h[n]; }
}

// ---------------------------------------------------------------------------
// Scan phase 2: inter-chunk recurrence per (b,d,n).
//   Hinit[c] = h_in;  h_in = P[c]*h_in + H[c]
// ---------------------------------------------------------------------------
__global__ void S4_scan_phase2(const float* __restrict__ P,
                               const float* __restrict__ H,
                               float* __restrict__ Hinit) {
  const int idx = blockIdx.x * blockDim.x + threadIdx.x;   // B*D*N = 65536
  const int n = idx & (N_ST - 1);
  const int d = (idx >> 4) & (D_SZ - 1);
  const int b = idx >> 15;
  const size_t base = ((size_t)b * D_SZ + d) * NCHUNK * N_ST + n;
  float hp = 0.0f;
#pragma unroll
  for (int c = 0; c < NCHUNK; ++c) {
    const size_t o = base + (size_t)c * N_ST;
    Hinit[o] = hp;
    hp = __builtin_fmaf(P[o], hp, H[o]);
  }
}

// ---------------------------------------------------------------------------
// Scan phase 3: re-scan each chunk seeded with Hinit; emit y.
//   y[m,d] = sum_n h[n]*C_t[n] + x[m,d]*Dvec[d]
// B and C tiles staged via async b128 -> LDS (2 per thread, one wait).
// ---------------------------------------------------------------------------
__global__ void S4_scan_phase3(const float* __restrict__ x,
                               const float* __restrict__ A,
                               const float* __restrict__ bcd,
                               const float* __restrict__ delta,
                               const float* __restrict__ Hinit,
                               const float* __restrict__ Dvec,
                               float* __restrict__ out) {
  __shared__ float sB[CHUNK * N_ST];       // B values
  __shared__ float sC[CHUNK * N_ST];       // C values
  const int idx = blockIdx.x * blockDim.x + threadIdx.x;
  const int d = idx & (D_SZ - 1);
  const int t = idx >> 11;
  const int b = t & (B_SZ - 1);
  const int c = t >> 1;

  const size_t rowBase = (size_t)b * L_SZ + (size_t)c * CHUNK;
  {
    const int l = threadIdx.x >> 2, q = threadIdx.x & 3;
    const float* gB = bcd + (rowBase + l) * E_SZ + R_SZ + q * 4;
    const float* gC = gB + N_ST;
    const unsigned lB = (unsigned)(uintptr_t)(sB + l * N_ST + q * 4);
    const unsigned lC = (unsigned)(uintptr_t)(sC + l * N_ST + q * 4);
    async_g2lds_b128(gB, lB);
    async_g2lds_b128(gC, lC);
    wait_async0();
  }
  __syncthreads();

  float Ar[N_ST];
#pragma unroll
  for (int n = 0; n < N_ST; ++n) Ar[n] = A[d * N_ST + n];

  const size_t pidx = (((size_t)b * D_SZ + d) * NCHUNK + c) * N_ST;
  float h[N_ST];
#pragma unroll
  for (int n = 0; n < N_ST; ++n) h[n] = Hinit[pidx + n];

  const float dskip = Dvec[d];

  for (int li = 0; li < CHUNK; ++li) {
    const size_t off = (rowBase + li) * D_SZ + d;
    float dlt = delta[off];
    float xv  = x[off];
    __builtin_prefetch(delta + off + D_SZ, 0, 0);
    float dx = dlt * xv;
    float y = 0.0f;
#pragma unroll
    for (int n = 0; n < N_ST; ++n) {
      float a = __expf(dlt * Ar[n]);
      h[n] = __builtin_fmaf(a, h[n], dx * sB[(li << 4) + n]);
      y = __builtin_fmaf(h[n], sC[(li << 4) + n], y);
    }
    out[off] = __builtin_fmaf(xv, dskip, y);
  }
}

// ---------------------------------------------------------------------------
// Host launch
// ---------------------------------------------------------------------------
extern "C" void kernel_launch(void* const* d_in, const int* in_sizes, int n_in,
                              void* d_out, int out_size, void* d_ws, size_t ws_size,
                              hipStream_t stream) {
  const float* x     = (const float*)d_in[0];   // [B,L,D]
  const float* Wxp   = (const float*)d_in[1];   // [160, D]
  const float* Wdt   = (const float*)d_in[2];   // [D, 128]
  const float* bdt   = (const float*)d_in[3];   // [D]
  const float* A     = (const float*)d_in[4];   // [D, N]
  const float* Dvec  = (const float*)d_in[5];   // [D]
  float* out = (float*)d_out;

  float* ws    = (float*)d_ws;
  float* bcd   = ws;                                   //  4096*160   = 655,360
  float* delta = bcd + (size_t)M_SZ * E_SZ;            //  4096*2048  = 8,388,608
  float* P     = delta + (size_t)M_SZ * D_SZ;          //  2*2048*32*16
  float* H     = P + (size_t)B_SZ * D_SZ * NCHUNK * N_ST;
  float* Hinit = H + (size_t)B_SZ * D_SZ * NCHUNK * N_ST;

  // 1) bcd = x . W_xproj^T  : 2560 tiles, 8 waves/block
  S4_gemm_xproj<<<320, 256, 0, stream>>>(x, Wxp, bcd);
  // 2) delta = softplus(dt . W_dt^T + b) : 32768 tiles
  S4_gemm_dt<<<4096, 256, 0, stream>>>(bcd, Wdt, bdt, delta);
  // 3) chunk-local scans : B*D*NCHUNK = 131072 threads
  S4_scan_phase1<<<512, 256, 0, stream>>>(x, A, bcd, delta, P, H);
  // 4) inter-chunk recurrence : B*D*N = 65536 threads
  S4_scan_phase2<<<256, 256, 0, stream>>>(P, H, Hinit);
  // 5) final scan + output : 131072 threads
  S4_scan_phase3<<<512, 256, 0, stream>>>(x, A, bcd, delta, Hinit, Dvec, out);
}